// YOSOHead_56083682951734
// MI455X (gfx1250) — compile-verified
//
#include <hip/hip_runtime.h>
#include <hip/hip_bf16.h>

// ===========================================================================
// YOSO head forward on gfx1250 (MI455X), exact fp32 via V_WMMA_F32_16X16X4_F32.
//
// Dims: B=4, N=100, C=HID=256, HW=25600, FF=2048, NH=8, DH=32.
//
// Input ordering assumption: harness flattens setup_inputs() as a JAX pytree
// (dict keys sorted alphabetically at each level):
//  0 features(4,256,160,160) 1 mask_preds(4,100,160,160)
//  2..7  cls_blocks[0].{W,b,g}, cls_blocks[1].{W,b,g}
//  8..11 f_atten.{Wl,b,bl,g}  12 f_norm_b 13 f_norm_g
// 14 fc_cls_W 15 fc_cls_b 16 fc_mask_W 17 fc_mask_b
// 18 ffn_W1 19 ffn_W2 20 ffn_b1 21 ffn_b2 22 ffn_norm_b 23 ffn_norm_g
// 24..27 k_atten.{Wl,b,bl,g} 28 k_norm_b 29 k_norm_g
// 30..35 mask_blocks[0].{W,b,g}, mask_blocks[1].{W,b,g}
// 36 s_atten.Wo 37 s_atten.Wqkv 38 s_atten.bo 39 s_atten.bqkv
// 40 s_norm_b 41 s_norm_g 42 proposal_kernels(4,100,256) 43 train_flag
//
// Output: [cls_score (4*100*3) | new_mask_preds (4*100*25600) | obj (4*100*256)]
// ===========================================================================

typedef float v2f __attribute__((ext_vector_type(2)));
typedef float v8f __attribute__((ext_vector_type(8)));

#define GF_RELU  1
#define GF_WT    2   // W stored K-major: B(k,n) = W[k*wLd + n]
#define GF_HARDA 4   // A element = (x > 0) ? 1 : 0

// Generic fp32 GEMM: Y[b,m,n] = act( sum_k A[b,m,k] * Bmat(k,n) + bias[n] )
//   A: X + b*xBatch + m*xRow + k            (fp32, row-major)
//   Bmat(k,n) = WT ? W[b][k*wLd+n] : W[b][n*wLd+k]
// One wave (32 threads) computes a 32x32 tile with 4 WMMA accumulators.
// Optional split-K: block z = b*nSplit + ks; partial written at Y + ks*ySplit.
template <int FLAGS>
__global__ __launch_bounds__(32) void gemm32_wmma(
    const float* __restrict__ X, long xRow, long xBatch, int M,
    const float* __restrict__ W, long wLd, long wBatch, int Nout,
    const float* __restrict__ bias,
    float* __restrict__ Y, long yRow, long yBatch, long ySplit,
    int K, int kChunk, int nSplit)
{
    constexpr bool WT   = (FLAGS & GF_WT)    != 0;
    constexpr bool HA   = (FLAGS & GF_HARDA) != 0;
    constexpr bool RELU = (FLAGS & GF_RELU)  != 0;

    const int lane = threadIdx.x;
    const int half = lane >> 4;       // 0: lanes 0-15, 1: lanes 16-31
    const int l15  = lane & 15;

    const int m0 = blockIdx.x * 32;
    const int n0 = blockIdx.y * 32;
    const int z  = blockIdx.z;
    const int ks = z % nSplit;
    const int b  = z / nSplit;

    // clamped load indices (rows/cols beyond M/Nout compute garbage, never stored)
    const int mA = min(m0 + l15, M - 1);
    const int mB = min(m0 + 16 + l15, M - 1);
    const int nA = min(n0 + l15, Nout - 1);
    const int nB = min(n0 + 16 + l15, Nout - 1);

    const float* __restrict__ xb = X + (long)b * xBatch;
    const float* __restrict__ x0 = xb + (long)mA * xRow;
    const float* __restrict__ x1 = xb + (long)mB * xRow;
    const float* __restrict__ wB = W + (long)b * wBatch;
    const float* __restrict__ wpA = WT ? wB : (wB + (long)nA * wLd);
    const float* __restrict__ wpB = WT ? wB : (wB + (long)nB * wLd);

    v8f a00 = {0,0,0,0,0,0,0,0};
    v8f a01 = a00, a10 = a00, a11 = a00;

    const int kbeg = ks * kChunk;
    int kend = kbeg + kChunk; if (kend > K) kend = K;

    for (int k = kbeg; k < kend; k += 4) {
        const int ka = k + 2 * half;   // K index of this lane's element pair
        v2f A0, A1, B0, B1;
        A0.x = x0[ka]; A0.y = x0[ka + 1];
        A1.x = x1[ka]; A1.y = x1[ka + 1];
        if (HA) {
            A0.x = A0.x > 0.f ? 1.f : 0.f;
            A0.y = A0.y > 0.f ? 1.f : 0.f;
            A1.x = A1.x > 0.f ? 1.f : 0.f;
            A1.y = A1.y > 0.f ? 1.f : 0.f;
        }
        if (WT) {
            const float* wr0 = wB + (long)ka * wLd;
            const float* wr1 = wr0 + wLd;
            B0.x = wr0[nA]; B0.y = wr1[nA];
            B1.x = wr0[nB]; B1.y = wr1[nB];
        } else {
            B0.x = wpA[ka]; B0.y = wpA[ka + 1];
            B1.x = wpB[ka]; B1.y = wpB[ka + 1];
        }
        a00 = __builtin_amdgcn_wmma_f32_16x16x4_f32(false, A0, false, B0, (short)0, a00, false, false);
        a01 = __builtin_amdgcn_wmma_f32_16x16x4_f32(false, A0, false, B1, (short)0, a01, false, false);
        a10 = __builtin_amdgcn_wmma_f32_16x16x4_f32(false, A1, false, B0, (short)0, a10, false, false);
        a11 = __builtin_amdgcn_wmma_f32_16x16x4_f32(false, A1, false, B1, (short)0, a11, false, false);
    }

    float* __restrict__ yp = Y + (long)b * yBatch + (long)ks * ySplit;
    const float bv0 = bias ? bias[nA] : 0.f;
    const float bv1 = bias ? bias[nB] : 0.f;
    const int col0 = n0 + l15;
    const int col1 = n0 + 16 + l15;
#pragma unroll
    for (int r = 0; r < 8; ++r) {
        const int row0 = m0 + r + 8 * half;   // C/D: VGPR r -> row r (+8 for hi half)
        const int row1 = row0 + 16;
        if (row0 < M) {
            if (col0 < Nout) { float v = a00[r] + bv0; if (RELU) v = fmaxf(v, 0.f); yp[(long)row0 * yRow + col0] = v; }
            if (col1 < Nout) { float v = a01[r] + bv1; if (RELU) v = fmaxf(v, 0.f); yp[(long)row0 * yRow + col1] = v; }
        }
        if (row1 < M) {
            if (col0 < Nout) { float v = a10[r] + bv0; if (RELU) v = fmaxf(v, 0.f); yp[(long)row1 * yRow + col0] = v; }
            if (col1 < Nout) { float v = a11[r] + bv1; if (RELU) v = fmaxf(v, 0.f); yp[(long)row1 * yRow + col1] = v; }
        }
    }
}

// Sum split-K partials: out[i] = sum_p P[p*stride + i]
__global__ void reduce_sum_k(const float* __restrict__ P, float* __restrict__ out,
                             int n, int parts, long stride)
{
    int i = blockIdx.x * 256 + threadIdx.x;
    if (i < n) {
        float s = 0.f;
        for (int p = 0; p < parts; ++p) s += P[(long)p * stride + i];
        out[i] = s;
    }
}

// Depthwise conv (KS=3, pad=1) over channel axis + ReLU.
// out[b,m,c] = relu( wd0*v[c-1] + wd1*v[c] + wd2*v[c+1] ), wd = wbuf[b,m,0..2]
__global__ __launch_bounds__(256) void dw_relu_k(const float* __restrict__ wbuf,
                                                 const float* __restrict__ val,
                                                 float* __restrict__ out)
{
    int bm = blockIdx.x;                 // 0..B*N-1
    int c  = threadIdx.x;                // 0..255
    const float* w = wbuf + (long)bm * 112;
    const float* v = val  + (long)bm * 256;
    float left  = (c > 0)   ? v[c - 1] : 0.f;
    float mid   = v[c];
    float right = (c < 255) ? v[c + 1] : 0.f;
    float o = w[0] * left + w[1] * mid + w[2] * right;
    out[(long)bm * 256 + c] = o > 0.f ? o : 0.f;
}

// LayerNorm over 256 channels; out = LN(in (+ addin)) * g + b, optional ReLU.
__global__ __launch_bounds__(256) void ln256_k(const float* __restrict__ in,
                                               const float* __restrict__ addin,
                                               const float* __restrict__ g,
                                               const float* __restrict__ bta,
                                               float* __restrict__ out, int relu)
{
    __shared__ float red[256];
    int row = blockIdx.x, c = threadIdx.x;
    long idx = (long)row * 256 + c;
    float x = in[idx];
    if (addin) x += addin[idx];
    red[c] = x; __syncthreads();
    for (int s = 128; s > 0; s >>= 1) { if (c < s) red[c] += red[c + s]; __syncthreads(); }
    float mean = red[0] * (1.f / 256.f);
    __syncthreads();
    float d = x - mean;
    red[c] = d * d; __syncthreads();
    for (int s = 128; s > 0; s >>= 1) { if (c < s) red[c] += red[c + s]; __syncthreads(); }
    float var = red[0] * (1.f / 256.f);
    float y = d * rsqrtf(var + 1e-5f) * g[c] + bta[c];
    if (relu) y = fmaxf(y, 0.f);
    out[idx] = y;
}

// MHA core: one block per (b, h, q). qkv layout (B, N, 768) = [q|k|v].
__global__ __launch_bounds__(128) void attn_k(const float* __restrict__ qkv,
                                              float* __restrict__ out)
{
    __shared__ float qs[32];
    __shared__ float sc[128];
    __shared__ float pe[128];
    int t = threadIdx.x;
    int idx = blockIdx.x;
    int q = idx % 100;
    int h = (idx / 100) & 7;
    int b = idx / 800;
    const float* base = qkv + (long)b * 100 * 768;
    if (t < 32) qs[t] = base[(long)q * 768 + h * 32 + t];
    __syncthreads();
    float s = -3.0e38f;
    if (t < 100) {
        const float* kr = base + (long)t * 768 + 256 + h * 32;
        float d = 0.f;
#pragma unroll
        for (int i = 0; i < 32; ++i) d += qs[i] * kr[i];
        s = d * 0.17677669529663687f;   // 1/sqrt(32)
    }
    sc[t] = s; __syncthreads();
    for (int st = 64; st > 0; st >>= 1) { if (t < st) sc[t] = fmaxf(sc[t], sc[t + st]); __syncthreads(); }
    float mx = sc[0]; __syncthreads();
    float e = (t < 100) ? __expf(s - mx) : 0.f;
    pe[t] = e; sc[t] = e; __syncthreads();
    for (int st = 64; st > 0; st >>= 1) { if (t < st) sc[t] += sc[t + st]; __syncthreads(); }
    float inv = 1.f / sc[0];
    __syncthreads();
    if (t < 32) {
        float o = 0.f;
        const float* vr = base + 512 + h * 32 + t;
        for (int k2 = 0; k2 < 100; ++k2) o += pe[k2] * vr[(long)k2 * 768];
        out[((long)b * 100 + q) * 256 + h * 32 + t] = o * inv;
    }
}

__global__ void copy_f_k(const float* __restrict__ a, float* __restrict__ o, int n)
{
    int i = blockIdx.x * 256 + threadIdx.x;
    if (i < n) o[i] = a[i];
}

extern "C" void kernel_launch(void* const* d_in, const int* in_sizes, int n_in,
                              void* d_out, int out_size, void* d_ws, size_t ws_size,
                              hipStream_t stream)
{
    (void)in_sizes; (void)n_in; (void)out_size;

    const float* features = (const float*)d_in[0];
    const float* maskpred = (const float*)d_in[1];
    const float* clsW0 = (const float*)d_in[2];
    const float* clsB0 = (const float*)d_in[3];
    const float* clsG0 = (const float*)d_in[4];
    const float* clsW1 = (const float*)d_in[5];
    const float* clsB1 = (const float*)d_in[6];
    const float* clsG1 = (const float*)d_in[7];
    const float* fA_Wl = (const float*)d_in[8];
    const float* fA_b  = (const float*)d_in[9];
    const float* fA_bl = (const float*)d_in[10];
    const float* fA_g  = (const float*)d_in[11];
    const float* fN_b  = (const float*)d_in[12];
    const float* fN_g  = (const float*)d_in[13];
    const float* fcClsW = (const float*)d_in[14];
    const float* fcClsB = (const float*)d_in[15];
    const float* fcMskW = (const float*)d_in[16];
    const float* fcMskB = (const float*)d_in[17];
    const float* W1     = (const float*)d_in[18];
    const float* W2     = (const float*)d_in[19];
    const float* b1v    = (const float*)d_in[20];
    const float* b2v    = (const float*)d_in[21];
    const float* ffnN_b = (const float*)d_in[22];
    const float* ffnN_g = (const float*)d_in[23];
    const float* kA_Wl = (const float*)d_in[24];
    const float* kA_b  = (const float*)d_in[25];
    const float* kA_bl = (const float*)d_in[26];
    const float* kA_g  = (const float*)d_in[27];
    const float* kN_b  = (const float*)d_in[28];
    const float* kN_g  = (const float*)d_in[29];
    const float* mskW0 = (const float*)d_in[30];
    const float* mskB0 = (const float*)d_in[31];
    const float* mskG0 = (const float*)d_in[32];
    const float* mskW1 = (const float*)d_in[33];
    const float* mskB1 = (const float*)d_in[34];
    const float* mskG1 = (const float*)d_in[35];
    const float* Wo    = (const float*)d_in[36];
    const float* Wqkv  = (const float*)d_in[37];
    const float* bo    = (const float*)d_in[38];
    const float* bqkv  = (const float*)d_in[39];
    const float* sN_b  = (const float*)d_in[40];
    const float* sN_g  = (const float*)d_in[41];
    const float* prop  = (const float*)d_in[42];
    // d_in[43] = train_flag (1 in this harness; cls head computed unconditionally)

    float* out = (float*)d_out;
    float* ws  = (float*)d_ws;

    const long SZ = 102400;                 // B*N*C
    long off = 0;
    float* f0   = ws + off; off += SZ;
    float* wb   = ws + off; off += 4 * 100 * 112;    // dysep linear out (ld=112)
    float* dep  = ws + off; off += SZ;               // depthwise+relu out
    float* dy   = ws + off; off += SZ;               // dysep output
    float* f1   = ws + off; off += SZ;
    float* k0b  = ws + off; off += SZ;
    float* qkvb = ws + off; off += 4 * 100 * 768;
    float* att  = ws + off; off += SZ;
    float* t1   = ws + off; off += SZ;
    float* t2   = ws + off; off += SZ;
    float* k1b  = ws + off; off += SZ;
    float* obj1 = ws + off; off += SZ;
    float* mk   = ws + off; off += SZ;
    float* hb   = ws + off; off += 4 * 100 * 2048;   // FFN hidden
    float* P    = ws + off;                          // split-K partials

    long remain = (long)(ws_size / 4) - off;
    int nsplit = 8;
    while (nsplit > 1 && (long)nsplit * SZ > remain) nsplit >>= 1;
    int kchunk = 25600 / nsplit;

    // ---- Big GEMM #1: f0[b,n,c] = sum_hw (mask>0) * features[b,c,hw] ----
    if (nsplit > 1) {
        gemm32_wmma<GF_HARDA><<<dim3(4, 8, 4 * nsplit), 32, 0, stream>>>(
            maskpred, 25600, 2560000L, 100,
            features, 25600, 6553600L, 256, nullptr,
            P, 256, 25600L, SZ, 25600, kchunk, nsplit);
        reduce_sum_k<<<(int)((SZ + 255) / 256), 256, 0, stream>>>(P, f0, (int)SZ, nsplit, SZ);
    } else {
        gemm32_wmma<GF_HARDA><<<dim3(4, 8, 4), 32, 0, stream>>>(
            maskpred, 25600, 2560000L, 100,
            features, 25600, 6553600L, 256, nullptr,
            f0, 256, 25600L, 0, 25600, 25600, 1);
    }

    // ---- dysep helper: dy = LN_g,b( wp @ relu(depthwise(value, wd)) ) ----
    auto dysep = [&](const float* Wl, const float* bl, const float* g, const float* bta,
                     const float* value) {
        // w = prop @ Wl^T + bl  -> wb (B,100,ld112), Nout=103
        gemm32_wmma<0><<<dim3(4, 4, 4), 32, 0, stream>>>(
            prop, 256, 25600L, 100, Wl, 256, 0, 103, bl,
            wb, 112, 11200L, 0, 256, 256, 1);
        dw_relu_k<<<400, 256, 0, stream>>>(wb, value, dep);
        // dy[m,c] = sum_n wp[m,n] * dep[n,c]  (wp = wb cols 3..102, K=100)
        gemm32_wmma<GF_WT><<<dim3(4, 8, 4), 32, 0, stream>>>(
            wb + 3, 112, 11200L, 100, dep, 256, 25600L, 256, nullptr,
            dy, 256, 25600L, 0, 100, 100, 1);
        ln256_k<<<400, 256, 0, stream>>>(dy, nullptr, g, bta, dy, 0);
    };

    dysep(fA_Wl, fA_bl, fA_g, fA_b, f0);
    ln256_k<<<400, 256, 0, stream>>>(f0, dy, fN_g, fN_b, f1, 0);     // f1 = LN(f0+dy)
    dysep(kA_Wl, kA_bl, kA_g, kA_b, f1);
    ln256_k<<<400, 256, 0, stream>>>(f1, dy, kN_g, kN_b, k0b, 0);    // k0 = LN(f1+dy)

    // ---- MHA ----
    gemm32_wmma<0><<<dim3(4, 24, 4), 32, 0, stream>>>(
        k0b, 256, 25600L, 100, Wqkv, 256, 0, 768, bqkv,
        qkvb, 768, 76800L, 0, 256, 256, 1);
    attn_k<<<3200, 128, 0, stream>>>(qkvb, att);
    gemm32_wmma<0><<<dim3(4, 8, 4), 32, 0, stream>>>(
        att, 256, 25600L, 100, Wo, 256, 0, 256, bo,
        t1, 256, 25600L, 0, 256, 256, 1);
    ln256_k<<<400, 256, 0, stream>>>(k0b, t1, sN_g, sN_b, k1b, 0);   // k1 = LN(k0+mha)

    // ---- FFN ----
    gemm32_wmma<GF_RELU><<<dim3(4, 64, 4), 32, 0, stream>>>(
        k1b, 256, 25600L, 100, W1, 256, 0, 2048, b1v,
        hb, 2048, 204800L, 0, 256, 256, 1);
    gemm32_wmma<0><<<dim3(4, 8, 4), 32, 0, stream>>>(
        hb, 2048, 204800L, 100, W2, 2048, 0, 256, b2v,
        t1, 256, 25600L, 0, 2048, 2048, 1);
    ln256_k<<<400, 256, 0, stream>>>(k1b, t1, ffnN_g, ffnN_b, obj1, 0);  // obj = LN(k1+ffn)

    // ---- cls head -> out[0..1200) ----
    gemm32_wmma<0><<<dim3(4, 8, 4), 32, 0, stream>>>(
        obj1, 256, 25600L, 100, clsW0, 256, 0, 256, nullptr,
        t1, 256, 25600L, 0, 256, 256, 1);
    ln256_k<<<400, 256, 0, stream>>>(t1, nullptr, clsG0, clsB0, t1, 1);
    gemm32_wmma<0><<<dim3(4, 8, 4), 32, 0, stream>>>(
        t1, 256, 25600L, 100, clsW1, 256, 0, 256, nullptr,
        t2, 256, 25600L, 0, 256, 256, 1);
    ln256_k<<<400, 256, 0, stream>>>(t2, nullptr, clsG1, clsB1, t2, 1);
    gemm32_wmma<0><<<dim3(4, 1, 4), 32, 0, stream>>>(
        t2, 256, 25600L, 100, fcClsW, 256, 0, 3, fcClsB,
        out, 3, 300L, 0, 256, 256, 1);

    // ---- mask head -> mask_kernels ----
    gemm32_wmma<0><<<dim3(4, 8, 4), 32, 0, stream>>>(
        obj1, 256, 25600L, 100, mskW0, 256, 0, 256, nullptr,
        t1, 256, 25600L, 0, 256, 256, 1);
    ln256_k<<<400, 256, 0, stream>>>(t1, nullptr, mskG0, mskB0, t1, 1);
    gemm32_wmma<0><<<dim3(4, 8, 4), 32, 0, stream>>>(
        t1, 256, 25600L, 100, mskW1, 256, 0, 256, nullptr,
        t2, 256, 25600L, 0, 256, 256, 1);
    ln256_k<<<400, 256, 0, stream>>>(t2, nullptr, mskG1, mskB1, t2, 1);
    gemm32_wmma<0><<<dim3(4, 8, 4), 32, 0, stream>>>(
        t2, 256, 25600L, 100, fcMskW, 256, 0, 256, fcMskB,
        mk, 256, 25600L, 0, 256, 256, 1);

    // ---- Big GEMM #2: new_mask_preds[b,q,hw] = mk[b,q,:] @ features[b,:,hw] ----
    gemm32_wmma<GF_WT><<<dim3(4, 800, 4), 32, 0, stream>>>(
        mk, 256, 25600L, 100,
        features, 25600, 6553600L, 25600, nullptr,
        out + 1200, 25600, 2560000L, 0, 256, 256, 1);

    // ---- obj_out = obj1 (flat layout identical) ----
    copy_f_k<<<400, 256, 0, stream>>>(obj1, out + 10241200, (int)SZ);
}